// Individual_39530878992727
// MI455X (gfx1250) — compile-verified
//
#include <hip/hip_runtime.h>
#include <math.h>

// ---------------------------------------------------------------------------
// Problem constants (from the reference)
// ---------------------------------------------------------------------------
#define INPUT_DIM   256
#define OUTPUT_DIM  16
#define NUM_MID     272            // 17 tiles of 16
#define NUM_NODES   528            // 132 K-chunks of 4
#define BATCH       32768
#define NT          17             // node tiles of 16
#define NKC         (NUM_NODES/4)  // 132 K-chunks in swizzled weight buffer
#define ROW_STRIDE  276            // LDS row pitch (floats); 276 mod 64 = 20 -> conflict-free b64 frag loads
#define PRE_STRIDE  17
#define WAVES       8
#define ROWS_PER_WAVE  16
#define ROWS_PER_BLOCK (WAVES * ROWS_PER_WAVE)   // 128

typedef float v2f  __attribute__((ext_vector_type(2)));
typedef float v8f  __attribute__((ext_vector_type(8)));
typedef unsigned int u32x4 __attribute__((ext_vector_type(4)));
typedef int   i32x4 __attribute__((ext_vector_type(4)));
typedef int   i32x8 __attribute__((ext_vector_type(8)));

// ---------------------------------------------------------------------------
// Prologue: build eff_w = weight * connectivity, pre-swizzled into the WMMA
// B-fragment layout:  wswz[kk_chunk][tile][lane][2]
//   element (K = kk*4 + 2*(lane/16) + b,  N = tile*16 + lane%16)
// so each 16x16x4 B-fragment is a contiguous 256B block -> one b64/lane.
// ---------------------------------------------------------------------------
__global__ void mask_weights_swz_kernel(const float* __restrict__ w,
                                        const int*   __restrict__ conn,
                                        float*       __restrict__ wswz, int n) {
  int o = blockIdx.x * blockDim.x + threadIdx.x;
  if (o >= n) return;
  int b    = o & 1;
  int lane = (o >> 1) & 31;
  int blk  = o >> 6;               // = kk_chunk*17 + tile
  int t    = blk % NT;
  int kkc  = blk / NT;
  int k    = kkc * 4 + ((lane >> 4) << 1) + b;
  int nn   = t * 16 + (lane & 15);
  int src  = k * NUM_MID + nn;
  wswz[o]  = conn[src] ? w[src] : 0.0f;
}

// Fetch w_hid[K=k][N=n] back out of the swizzled layout (for the scalar
// intra-tile part).  k is the absolute node row (>= INPUT_DIM).
__device__ __forceinline__ float wswz_at(const float* wswz, int k, int n) {
  int kkc  = k >> 2;
  int kq   = k & 3;
  int lane = ((kq >> 1) << 4) + (n & 15);
  int o    = (((kkc * NT + (n >> 4)) << 5) + lane) * 2 + (kq & 1);
  return wswz[o];
}

// ---------------------------------------------------------------------------
// Main kernel: each wave owns 16 batch rows end-to-end.
//   Stage X tile via TENSOR_LOAD_TO_LDS (TDM pads each 256-DWORD row by 20
//   DWORDs -> LDS pitch 276 floats, matching ROW_STRIDE).
//   Phase A: base = Xtile @ eff_w[:256]      (v_wmma_f32_16x16x4_f32)
//   Phase B: blocked triangular recurrence (WMMA GEMM + sequential 16-node
//   intra-tile part).
// ---------------------------------------------------------------------------
__global__ __launch_bounds__(256)
void Individual_39530878992727_kernel(const float* __restrict__ x,
                                      const float* __restrict__ wswz,
                                      const float* __restrict__ bias,
                                      const int*   __restrict__ exist,
                                      float*       __restrict__ out)
{
  __shared__ __align__(16) float lds_res[WAVES][ROWS_PER_WAVE * ROW_STRIDE]; // X tile, then res buffer
  __shared__ __align__(16) float lds_pre[WAVES][ROWS_PER_WAVE * PRE_STRIDE]; // pre-activation spill

  const int tid  = threadIdx.x;
  const int wave = tid >> 5;
  const int lane = tid & 31;
  const int lrow = lane & 15;   // A: M row / B,D: N col
  const int lhi  = lane >> 4;   // half-wave select
  const int kcol = 2 * lhi;     // K offset within a K=4 chunk
  const int row0 = blockIdx.x * ROWS_PER_BLOCK + wave * ROWS_PER_WAVE;

  float* resw = lds_res[wave];
  float* prew = lds_pre[wave];

  __builtin_prefetch(wswz, 0, 1);           // global_prefetch_b8 (L2 warm)

  // ---- Stage X tile (16 x 256 f32) into LDS via the Tensor Data Mover ----
  // D# group0: count=1 | lds_addr | global_addr(57b) | type=2
  // D# group1: data_size=4B, pad_enable, pad_interval=7 (256 DW),
  //            pad_amount=19 (+20 DW) => LDS row pitch 276 floats.
  //            tensor_dim0=256, tensor_dim1=BATCH, tile=256x16, stride0=256.
  {
    unsigned long long gaddr = (unsigned long long)(const void*)(x + (size_t)row0 * INPUT_DIM);
    unsigned int lds_off = (unsigned int)(unsigned long long)(const void*)resw; // low 32b of generic ptr = LDS offset

    u32x4 g0;
    g0[0] = 1u;                                   // count=1 (valid descriptor)
    g0[1] = lds_off;                              // lds_addr (bytes)
    g0[2] = (unsigned int)gaddr;                  // global_addr[31:0]
    g0[3] = (unsigned int)((gaddr >> 32) & 0x01FFFFFFu) | 0x80000000u; // addr[56:32] | type=2

    i32x8 g1;
    g1[0] = (int)((2u << 16) | (1u << 20) | (7u << 22) | (19u << 25)); // 4B elems, pad 256DW->+20DW
    g1[1] = (int)(256u << 16);                    // tensor_dim0 = 256
    g1[2] = (int)(32768u << 16);                  // tensor_dim1 = 32768 (lo16)
    g1[3] = (int)(256u << 16);                    // tensor_dim1 hi=0 | tile_dim0 = 256
    g1[4] = 16;                                   // tile_dim1 = 16, tile_dim2 = 0
    g1[5] = 256;                                  // tensor_dim0_stride = 256
    g1[6] = 0;
    g1[7] = 0;

    i32x4 gz4 = {0, 0, 0, 0};
    i32x8 gz8 = {0, 0, 0, 0, 0, 0, 0, 0};
    __builtin_amdgcn_tensor_load_to_lds(g0, g1, gz4, gz4, gz8, 0);
    __builtin_amdgcn_s_wait_tensorcnt(0);
  }
  __syncthreads();

  // ---- Phase A: base accumulation, 17 live 16x16 f32 accumulators ----
  v8f acc[NT];
  #pragma unroll
  for (int t = 0; t < NT; ++t) acc[t] = (v8f){0.f,0.f,0.f,0.f,0.f,0.f,0.f,0.f};

  const float* wlane = wswz + lane * 2;          // per-lane base into swizzled weights

  for (int kk = 0; kk < INPUT_DIM / 4; ++kk) {   // 64 K-steps of 4
    const v2f a = *(const v2f*)(resw + lrow * ROW_STRIDE + kk * 4 + kcol);
    const float* wk = wlane + (size_t)kk * (NT * 64);
    #pragma unroll
    for (int t = 0; t < NT; ++t) {
      const v2f b = *(const v2f*)(wk + t * 64);  // one b64, immediate offset
      acc[t] = __builtin_amdgcn_wmma_f32_16x16x4_f32(
          false, a, false, b, (short)0, acc[t], false, false);
    }
  }
  __syncthreads();   // done reading X tile; resw becomes the res buffer

  // ---- Phase B: blocked triangular recurrence over 17 node tiles ----
  const float* whid_swz = wswz + (size_t)(INPUT_DIM / 4) * (NT * 64); // K-chunks >= 64
  #pragma unroll
  for (int t = 0; t < NT; ++t) {
    v8f c = acc[t];

    // GEMM part: pre_tile += res[:, 0:16t] @ w_hid[0:16t, tile]
    const float* wt = whid_swz + lane * 2 + t * 64;
    for (int kk = 0; kk < t * 4; ++kk) {
      const v2f a = *(const v2f*)(resw + lrow * ROW_STRIDE + kk * 4 + kcol);
      const v2f b = *(const v2f*)(wt + (size_t)kk * (NT * 64));
      c = __builtin_amdgcn_wmma_f32_16x16x4_f32(
          false, a, false, b, (short)0, c, false, false);
    }

    // Spill D fragment (16x16 pre-activations) to per-wave LDS scratch
    #pragma unroll
    for (int r = 0; r < 8; ++r)
      prew[(r + 8 * lhi) * PRE_STRIDE + lrow] = c[r];

    // Intra-tile sequential part: lane m (0..15) owns batch row m.
    // Per-wave LDS ops are in-order, so the spill above is visible here.
    if (lane < 16) {
      const int m = lane;
      for (int j = 0; j < 16; ++j) {
        const int node = t * 16 + j;
        float p = prew[m * PRE_STRIDE + j] + bias[node];
        for (int j2 = 0; j2 < j; ++j2)
          p += resw[m * ROW_STRIDE + t * 16 + j2] *
               wswz_at(wswz, INPUT_DIM + t * 16 + j2, node);
        float s = (float)exist[node] * (1.0f / (1.0f + __expf(-p)));
        resw[m * ROW_STRIDE + node] = s;   // future tiles' A-frags read this
      }
    }
  }

  // ---- Epilogue: write res[:, NUM_MID-16 : NUM_MID] (16 rows x 16 cols) ----
  #pragma unroll
  for (int i = 0; i < 8; ++i) {
    int flat = i * 32 + lane;              // 0..255
    int r  = flat >> 4;
    int cI = flat & 15;
    out[(size_t)(row0 + r) * OUTPUT_DIM + cI] =
        resw[r * ROW_STRIDE + (NUM_MID - OUTPUT_DIM) + cI];
  }
}

// ---------------------------------------------------------------------------
// Launch: d_in = {x, weight, bias, connectivity, node_existence}
// ---------------------------------------------------------------------------
extern "C" void kernel_launch(void* const* d_in, const int* in_sizes, int n_in,
                              void* d_out, int out_size, void* d_ws, size_t ws_size,
                              hipStream_t stream) {
  const float* x     = (const float*)d_in[0];
  const float* w     = (const float*)d_in[1];
  const float* bias  = (const float*)d_in[2];
  const int*   conn  = (const int*)d_in[3];
  const int*   exist = (const int*)d_in[4];
  float* wswz = (float*)d_ws;                     // 528*272*4 = 574 KB scratch
  float* outp = (float*)d_out;

  const int nW = NUM_NODES * NUM_MID;
  mask_weights_swz_kernel<<<(nW + 255) / 256, 256, 0, stream>>>(w, conn, wswz, nW);

  Individual_39530878992727_kernel<<<BATCH / ROWS_PER_BLOCK, 256, 0, stream>>>(
      x, wswz, bias, exist, outp);
}